// hetero_conv_55422257988272
// MI455X (gfx1250) — compile-verified
//
#include <hip/hip_runtime.h>

#define N_NODES 100000
#define N_EDGES 400000
#define N_ETYPES 3
#define N_LAYERS 3
#define D 64
#define LDS_TSTRIDE 68          // transposed [n][k] row stride: bank = (4n+k)%64,
                                // half-wave K pairs {kb,kb+1} vs {kb+2,kb+3} are
                                // disjoint across all lanes -> conflict-free b64
#define MAT_FLTS (64 * LDS_TSTRIDE)

typedef float v2f __attribute__((ext_vector_type(2)));
typedef float v8f __attribute__((ext_vector_type(8)));

// ---------------------------------------------------------------- zero fill
__global__ void zero_kernel(float* __restrict__ p, int n) {
    int i = blockIdx.x * blockDim.x + threadIdx.x;
    if (i < n) p[i] = 0.0f;
}

// ---------------------------------------------------------------- degree
__global__ void degree_kernel(const int* __restrict__ dst,
                              float* __restrict__ deg, int E) {
    int e = blockIdx.x * blockDim.x + threadIdx.x;
    if (e < E) atomicAdd(&deg[dst[e]], 1.0f);
}

// ------------------------------------------------- gather + scatter-add, all 3 etypes
// blockIdx.y = etype. 16 threads/edge, 4 floats each. h/agg are L2-resident
// (25.6 MB per N x 64 fp32 matrix vs 192 MB L2), so atomics resolve in L2.
__global__ void scatter3_kernel(const float* __restrict__ h,
                                const int* __restrict__ srcAll,
                                const int* __restrict__ dstAll,
                                float* __restrict__ aggAll, int E, int N) {
    const int t = blockIdx.y;
    const int* src = srcAll + (size_t)t * E;
    const int* dst = dstAll + (size_t)t * E;
    float* agg = aggAll + (size_t)t * N * D;
    int tid = blockIdx.x * blockDim.x + threadIdx.x;
    int e = tid >> 4;
    if (e >= E) return;
    int c = (tid & 15) * 4;
    const float4 v = *(const float4*)(h + (size_t)src[e] * D + c);
    float* a = agg + (size_t)dst[e] * D + c;
    atomicAdd(a + 0, v.x);
    atomicAdd(a + 1, v.y);
    atomicAdd(a + 2, v.z);
    atomicAdd(a + 3, v.w);
}

// ------------------------------------------------- fused per-layer kernel
// out[m,:] = sum_t act( h[m,:] @ Wself[t] + mean_t[m,:] @ Wneigh[t] + b[t] )
// - 6 weight matrices transpose-staged to LDS: B fragment = 1 contiguous,
//   conflict-free ds_load_b64 (no register shuffles on the WMMA chain)
// - bias async-staged (global_load_async_to_lds_b128 / ASYNCcnt)
// - h A-fragment cached in 16 v2f registers, reused across the 3 etypes
// - direct store, no RMW; `out` may alias `h` (each thread touches only its rows)
// - launch_bounds(128,1): LDS already caps occupancy at 3 blocks/WGP, so lift
//   the VGPR cap to keep hfrag/acc/sum resident (no scratch spills)
__global__ void __launch_bounds__(128, 1)
sage_layer_kernel(const float* h,
                  const float* __restrict__ aggAll,  // [3][N][64]
                  const float* __restrict__ degAll,  // [3][N]
                  const float* __restrict__ Wself3,  // [3][64][64]
                  const float* __restrict__ Wneigh3, // [3][64][64]
                  const float* __restrict__ bias3,   // [3][64]
                  float* out,
                  int N, int applyRelu) {
    __shared__ float ldsT[6 * MAT_FLTS + 192];   // 6 transposed mats + bias; ~105 KB
    float* ldsBias = &ldsT[6 * MAT_FLTS];

    const int tid     = threadIdx.x;       // 0..127 (4 waves)
    const int wave    = tid >> 5;
    const int lane    = tid & 31;
    const int halfSel = lane >> 4;         // 0: K pair {0,1}, 1: {2,3}
    const int l15     = lane & 15;
    const int row0    = blockIdx.x * 64 + wave * 16;

    // ---- transpose-stage 6 matrices: item = (mat, n, 4 k's). Column loads are
    // fully coalesced (consecutive n across lanes); b128 LDS store per item.
    for (int it = 0; it < 48; ++it) {
        int q   = it * 128 + tid;          // 0..6143
        int mat = q >> 10;                 // 0..5 (uniform per iteration)
        int rem = q & 1023;
        int n   = rem & 63;
        int k0  = (rem >> 6) * 4;          // 0,4,...,60
        const float* gsrc = (mat < 3) ? (Wself3 + (size_t)mat * 4096)
                                      : (Wneigh3 + (size_t)(mat - 3) * 4096);
        float4 v;
        v.x = gsrc[(k0 + 0) * 64 + n];
        v.y = gsrc[(k0 + 1) * 64 + n];
        v.z = gsrc[(k0 + 2) * 64 + n];
        v.w = gsrc[(k0 + 3) * 64 + n];
        *(float4*)&ldsT[mat * MAT_FLTS + n * LDS_TSTRIDE + k0] = v;
    }
    // ---- async-stage bias (3x64 floats = 48 x 16B chunks) via ASYNCcnt path
    if (tid < 48) {
        uint64_t ga   = (uint64_t)(uintptr_t)(bias3 + tid * 4);
        unsigned lofs = (unsigned)(uintptr_t)(&ldsBias[tid * 4]);
        asm volatile("global_load_async_to_lds_b128 %0, %1, off"
                     :: "v"(lofs), "v"(ga) : "memory");
    }
    asm volatile("s_wait_asynccnt 0x0" ::: "memory");
    __syncthreads();

    // A-matrix row for this lane (same row for both lane halves). Clamped tail
    // rows may read racy in-place data, but their stores are predicated off.
    int myRow = row0 + l15;
    int rowC  = myRow < N ? myRow : (N - 1);
    const float* hrow = h + (size_t)rowC * D;

    // Cache this lane's h fragment: 16 k-steps x v2f (reused for all etypes)
    v2f hfrag[16];
#pragma unroll
    for (int ks = 0; ks < 16; ++ks) {
        int kb = ks * 4 + halfSel * 2;
        hfrag[ks].x = hrow[kb];
        hfrag[ks].y = hrow[kb + 1];
    }

    v8f sum[4];
#pragma unroll
    for (int ct = 0; ct < 4; ++ct)
#pragma unroll
        for (int r = 0; r < 8; ++r) sum[ct][r] = 0.0f;

#pragma unroll 1
    for (int t = 0; t < N_ETYPES; ++t) {
        const float* arow = aggAll + ((size_t)t * N + rowC) * D;
        // v_rcp_f32 (~1 ulp) instead of the full v_div_* sequence
        const float invd = __builtin_amdgcn_rcpf(
            fmaxf(degAll[(size_t)t * N + rowC], 1.0f));
        const int bS = t * MAT_FLTS;
        const int bN = (3 + t) * MAT_FLTS;

        v8f acc[4];
#pragma unroll
        for (int ct = 0; ct < 4; ++ct) {
            float bv = ldsBias[t * 64 + ct * 16 + l15];
#pragma unroll
            for (int r = 0; r < 8; ++r) acc[ct][r] = bv;
        }

#pragma unroll
        for (int ks = 0; ks < 16; ++ks) {
            const int kb = ks * 4 + halfSel * 2;
            v2f a_m;
            a_m.x = arow[kb] * invd;
            a_m.y = arow[kb + 1] * invd;
#pragma unroll
            for (int ct = 0; ct < 4; ++ct) {
                const int n = ct * 16 + l15;
                v2f b_s = *(const v2f*)&ldsT[bS + n * LDS_TSTRIDE + kb];
                v2f b_m = *(const v2f*)&ldsT[bN + n * LDS_TSTRIDE + kb];
                acc[ct] = __builtin_amdgcn_wmma_f32_16x16x4_f32(
                    false, hfrag[ks], false, b_s, (short)0, acc[ct], false, false);
                acc[ct] = __builtin_amdgcn_wmma_f32_16x16x4_f32(
                    false, a_m, false, b_m, (short)0, acc[ct], false, false);
            }
        }

        // per-etype activation, then cross-relation sum
#pragma unroll
        for (int ct = 0; ct < 4; ++ct)
#pragma unroll
            for (int r = 0; r < 8; ++r) {
                float v = acc[ct][r];
                if (applyRelu) v = fmaxf(v, 0.0f);
                sum[ct][r] += v;
            }
    }

    // Direct store (no zero-init, no RMW). D layout: VGPR r -> rows r / r+8.
#pragma unroll
    for (int ct = 0; ct < 4; ++ct) {
        const int n = ct * 16 + l15;
#pragma unroll
        for (int r = 0; r < 8; ++r) {
            int m = row0 + r + halfSel * 8;
            if (m < N) out[(size_t)m * D + n] = sum[ct][r];
        }
    }
}

// ----------------------------------------------------------------------------
extern "C" void kernel_launch(void* const* d_in, const int* in_sizes, int n_in,
                              void* d_out, int out_size, void* d_ws, size_t ws_size,
                              hipStream_t stream) {
    const float* feat    = (const float*)d_in[0];                 // [N, 64]
    const float* W_self  = (const float*)d_in[1];                 // [3,3,64,64]
    const float* W_neigh = (const float*)d_in[2];                 // [3,3,64,64]
    const float* bias    = (const float*)d_in[3];                 // [3,3,64]
    const int*   src     = (const int*)d_in[4];                   // [3,E]
    const int*   dst     = (const int*)d_in[5];                   // [3,E]

    const int N = N_NODES, E = N_EDGES;
    const size_t ND = (size_t)N * D;

    float* hbuf = (float*)d_ws;         // feature buffer (in-place for layer 1)
    float* agg  = hbuf + ND;            // [3][N][64] per-etype aggregation
    float* deg  = agg + 3 * ND;         // [3][N] degrees (layer-invariant)

    // ---- degrees (recomputed every call for determinism)
    {
        int cnt = N_ETYPES * N;
        zero_kernel<<<(cnt + 255) / 256, 256, 0, stream>>>(deg, cnt);
        for (int t = 0; t < N_ETYPES; ++t)
            degree_kernel<<<(E + 255) / 256, 256, 0, stream>>>(
                dst + (size_t)t * E, deg + (size_t)t * N, E);
    }

    const float* hcur = feat;
    for (int l = 0; l < N_LAYERS; ++l) {
        // zero + scatter all 3 etype aggregations
        int aggCnt = (int)(3 * ND);
        zero_kernel<<<(aggCnt + 255) / 256, 256, 0, stream>>>(agg, aggCnt);
        dim3 sgrid((E * 16 + 255) / 256, N_ETYPES);
        scatter3_kernel<<<sgrid, 256, 0, stream>>>(hcur, src, dst, agg, E, N);

        // fused 3-etype GEMM + relu + cross-relation sum
        float* out = (l == N_LAYERS - 1) ? (float*)d_out : hbuf;
        const size_t wofs = (size_t)l * N_ETYPES * D * D;
        const size_t bofs = (size_t)l * N_ETYPES * D;
        sage_layer_kernel<<<(N + 63) / 64, 128, 0, stream>>>(
            hcur, agg, deg,
            W_self + wofs, W_neigh + wofs, bias + bofs,
            out, N, (l < N_LAYERS - 1) ? 1 : 0);
        hcur = out;   // layer0: feat->hbuf; layer1: in-place; layer2: ->d_out
    }
}